// DCTTransform_56075093016889
// MI455X (gfx1250) — compile-verified
//
#include <hip/hip_runtime.h>

typedef __attribute__((ext_vector_type(2))) float v2f;
typedef __attribute__((ext_vector_type(8))) float v8f;

#define NSZ   512
#define IMG   (512 * 512)
#define BATCH 96          // 32 batch * 3 channels

// ---------------------------------------------------------------------------
// Orthonormal DCT-II matrix: D[k][m] = cos(pi*(2m+1)*k/(2N)) * sqrt(2/N),
// row k=0 scaled by 1/sqrt(2).  sqrt(2/512) = 1/16 exactly.
// Angle reduced mod 2048 (cos(pi*t/1024) has period t=2048) so cosf stays
// accurate without double-precision argument reduction.
// ---------------------------------------------------------------------------
__global__ void gen_dct_mat(float* __restrict__ D, float* __restrict__ Dt) {
    int idx = blockIdx.x * blockDim.x + threadIdx.x;   // 0 .. 512*512-1
    int k = idx >> 9;
    int m = idx & (NSZ - 1);
    int t = ((2 * m + 1) * k) & 2047;                  // exact periodic reduction
    float ang = 3.14159265358979323846f * (float)t * (1.0f / 1024.0f);
    float scale = (k == 0) ? 0.04419417382415922f      // 1/(16*sqrt(2))
                           : 0.0625f;                  // 1/16
    float val = __cosf(ang) * scale;
    D[k * NSZ + m]  = val;
    Dt[m * NSZ + k] = val;
}

// ---------------------------------------------------------------------------
// Batched 512x512x512 GEMM, C = A @ B (optionally B elementwise-scaled by a
// shared per-image mask).  Each wave computes a 64x64 C block as a 4x4 grid
// of 16x16 V_WMMA_F32_16X16X4_F32 accumulators (128 acc VGPRs), so every
// A fragment is reused 4x and every B fragment 4x: 16 WMMAs per 12 loads.
//
// Fragment layouts (ISA 7.12.2, wave32):
//   A 16x4 :  lane L -> row M = L&15,  a[v] = A[M][k + 2*(L>>4) + v]
//   B 4x16 :  lane L -> col N = L&15,  b[v] = B[k + 2*(L>>4) + v][N]
//   C 16x16:  lane L -> col N = L&15,  c[r] = C[r + 8*(L>>4)][N]
// ---------------------------------------------------------------------------
template <bool MASKED>
__global__ __launch_bounds__(256)
void dct_gemm512(const float* __restrict__ A, long sA,
                 const float* __restrict__ B, long sB,
                 const float* __restrict__ Mk,
                 float* __restrict__ C, long sC) {
    const int lane = threadIdx.x & 31;
    const int wave = threadIdx.x >> 5;
    const int idx  = blockIdx.x * 8 + wave;   // 0..63 (8x8 grid of 64x64 blocks)
    const int tM   = idx >> 3;
    const int tN   = idx & 7;
    const int b    = blockIdx.z;
    const int l15  = lane & 15;
    const int half = lane >> 4;

    const float* Ab = A + (long)b * sA + (long)(tM * 64 + l15) * NSZ;
    const float* Bb = B + (long)b * sB + tN * 64 + l15;
    const float* Mb = Mk + tN * 64 + l15;      // mask shared across batch

    v8f acc[4][4];
#pragma unroll
    for (int mt = 0; mt < 4; ++mt)
#pragma unroll
        for (int nt = 0; nt < 4; ++nt)
            acc[mt][nt] = (v8f){};

#pragma unroll 2
    for (int k = 0; k < NSZ; k += 4) {
        const int ka = k + 2 * half;

        // A fragments: 4 x b64, one 16-row subtile each (reused across 4 N).
        v2f a[4];
#pragma unroll
        for (int mt = 0; mt < 4; ++mt)
            a[mt] = *(const v2f*)(Ab + (long)mt * 16 * NSZ + ka);

        // B fragments: 8 x b32 (rows ka, ka+1), reused across 4 M.
        v2f bb[4];
#pragma unroll
        for (int nt = 0; nt < 4; ++nt) {
            float b0 = Bb[(long)ka * NSZ + nt * 16];
            float b1 = Bb[(long)(ka + 1) * NSZ + nt * 16];
            if (MASKED) {
                b0 *= Mb[(long)ka * NSZ + nt * 16];
                b1 *= Mb[(long)(ka + 1) * NSZ + nt * 16];
            }
            bb[nt].x = b0;
            bb[nt].y = b1;
        }

        // 16 matrix ops per K-step.
#pragma unroll
        for (int mt = 0; mt < 4; ++mt)
#pragma unroll
            for (int nt = 0; nt < 4; ++nt)
                acc[mt][nt] = __builtin_amdgcn_wmma_f32_16x16x4_f32(
                    false, a[mt], false, bb[nt], (short)0, acc[mt][nt],
                    false, false);
    }

    float* Cb = C + (long)b * sC + (long)(tM * 64) * NSZ + tN * 64 + l15;
#pragma unroll
    for (int mt = 0; mt < 4; ++mt)
#pragma unroll
        for (int nt = 0; nt < 4; ++nt)
#pragma unroll
            for (int r = 0; r < 8; ++r)
                Cb[(long)(mt * 16 + r + 8 * half) * NSZ + nt * 16] =
                    acc[mt][nt][r];
}

// ---------------------------------------------------------------------------
// Pipeline:
//   T    = D  @ x           (forward, axis 0)
//   dct2 = T  @ D^T         (forward, axis 1)
//   for each output j (mask order LL,LH,HL,HH -> mask idx 3,0,1,2):
//     T     = D^T @ (mask_j ∘ dct2)   (mask fused into B load)
//     out_j = T   @ D
// ---------------------------------------------------------------------------
extern "C" void kernel_launch(void* const* d_in, const int* in_sizes, int n_in,
                              void* d_out, int out_size, void* d_ws, size_t ws_size,
                              hipStream_t stream) {
    const float* x     = (const float*)d_in[0];   // [32,3,512,512]
    const float* masks = (const float*)d_in[1];   // [4,512,512]
    float* out = (float*)d_out;                   // 4 x [32,3,512,512]

    float* D    = (float*)d_ws;                   // 1 MB
    float* Dt   = D + IMG;                        // 1 MB
    float* T    = Dt + IMG;                       // 96 MB (reused)
    float* dct2 = T + (long)BATCH * IMG;          // 96 MB

    gen_dct_mat<<<IMG / 256, 256, 0, stream>>>(D, Dt);

    dim3 grid(8, 1, BATCH);   // 8 blocks * 8 waves = 64 (64x64) blocks/image

    // T = D @ x
    dct_gemm512<false><<<grid, 256, 0, stream>>>(D, 0L, x, (long)IMG, masks, T, (long)IMG);
    // dct2 = T @ D^T   ((D^T)[k][n] == Dt[k][n])
    dct_gemm512<false><<<grid, 256, 0, stream>>>(T, (long)IMG, Dt, 0L, masks, dct2, (long)IMG);

    const int mask_for_out[4] = {3, 0, 1, 2};     // out order LL,LH,HL,HH
    for (int j = 0; j < 4; ++j) {
        const float* mk = masks + (long)mask_for_out[j] * IMG;
        // T = D^T @ (mask ∘ dct2)   ((D^T)[m][k] == Dt[m][k])
        dct_gemm512<true><<<grid, 256, 0, stream>>>(Dt, 0L, dct2, (long)IMG, mk, T, (long)IMG);
        // out_j = T @ D
        dct_gemm512<false><<<grid, 256, 0, stream>>>(T, (long)IMG, D, 0L, masks,
                                                     out + (long)j * BATCH * IMG, (long)IMG);
    }
}